// DensityVQC_18382460027033
// MI455X (gfx1250) — compile-verified
//
#include <hip/hip_runtime.h>

typedef __attribute__((ext_vector_type(2))) float v2f;
typedef __attribute__((ext_vector_type(8))) float v8f;

// ---------------------------------------------------------------------------
// Prologue: build 4x4 circuit unitary U, M = conj(U), and pre-swizzle the two
// constant WMMA *A* operands for V_WMMA_F32_16X16X4_F32.
//   A layout (f32 16x4, 2 VGPRs/lane): lane L -> row m = L&15,
//   vgpr j -> K = j + (L<16 ? 0 : 2).
//   A1 rows 0-3 = Mr, rows 4-7 = Mi, rows 8-15 = 0
//   A2 rows 0-3 = -Mi, rows 4-7 = Mr, rows 8-15 = 0
//   tab[0..31]  = A1 vgpr0, tab[32..63] = A1 vgpr1
//   tab[64..95] = A2 vgpr0, tab[96..127]= A2 vgpr1
// ---------------------------------------------------------------------------
__global__ __launch_bounds__(32) void vqc_build_tables(
    const float* __restrict__ ry, const float* __restrict__ rz,
    float* __restrict__ tab) {
  if (threadIdx.x != 0) return;

  float Ur[16], Ui[16];
  for (int i = 0; i < 16; ++i) { Ur[i] = (i % 5 == 0) ? 1.f : 0.f; Ui[i] = 0.f; }
  const int perm[4] = {0, 1, 3, 2};  // CNOT row permutation (swap rows 2,3)

  for (int l = 0; l < 6; ++l) {
    // RY(w0) kron RY(w1): real 4x4
    float h0 = 0.5f * ry[l * 2 + 0], h1 = 0.5f * ry[l * 2 + 1];
    float c0 = cosf(h0), s0 = sinf(h0), c1 = cosf(h1), s1 = sinf(h1);
    float R0[4] = {c0, -s0, s0, c0};
    float R1[4] = {c1, -s1, s1, c1};
    float F[16];
    for (int a = 0; a < 2; ++a)
      for (int b = 0; b < 2; ++b)
        for (int cc = 0; cc < 2; ++cc)
          for (int d = 0; d < 2; ++d)
            F[(2 * a + b) * 4 + (2 * cc + d)] = R0[a * 2 + cc] * R1[b * 2 + d];
    // RZ(w0) kron RZ(w1): diagonal phases exp(i*ph[j])
    float t0 = rz[l * 2 + 0], t1 = rz[l * 2 + 1];
    float ph[4];
    for (int j = 0; j < 4; ++j) {
      int a = j >> 1, b = j & 1;
      ph[j] = 0.5f * ((a ? t0 : -t0) + (b ? t1 : -t1));
    }
    // Layer L = CNOT * diag(exp(i*ph)) * F  (row permute of phased F)
    float Lr[16], Li[16];
    for (int i = 0; i < 4; ++i) {
      int j = perm[i];
      float cr = cosf(ph[j]), ci = sinf(ph[j]);
      for (int k = 0; k < 4; ++k) {
        Lr[i * 4 + k] = cr * F[j * 4 + k];
        Li[i * 4 + k] = ci * F[j * 4 + k];
      }
    }
    // U = L @ U  (complex 4x4)
    float Nr[16], Ni[16];
    for (int i = 0; i < 4; ++i)
      for (int k = 0; k < 4; ++k) {
        float ar = 0.f, ai = 0.f;
        for (int m = 0; m < 4; ++m) {
          ar += Lr[i * 4 + m] * Ur[m * 4 + k] - Li[i * 4 + m] * Ui[m * 4 + k];
          ai += Lr[i * 4 + m] * Ui[m * 4 + k] + Li[i * 4 + m] * Ur[m * 4 + k];
        }
        Nr[i * 4 + k] = ar;
        Ni[i * 4 + k] = ai;
      }
    for (int i = 0; i < 16; ++i) { Ur[i] = Nr[i]; Ui[i] = Ni[i]; }
  }

  // M = conj(U)
  float Mr[16], Mi[16];
  for (int i = 0; i < 16; ++i) { Mr[i] = Ur[i]; Mi[i] = -Ui[i]; }

  for (int j = 0; j < 2; ++j)
    for (int L = 0; L < 32; ++L) {
      int k = j + ((L < 16) ? 0 : 2);
      int m = L & 15;
      float a1, a2;
      if (m < 4)      { a1 = Mr[m * 4 + k];       a2 = -Mi[m * 4 + k]; }
      else if (m < 8) { a1 = Mi[(m - 4) * 4 + k]; a2 =  Mr[(m - 4) * 4 + k]; }
      else            { a1 = 0.f;                 a2 = 0.f; }
      tab[j * 32 + L]      = a1;
      tab[64 + j * 32 + L] = a2;
    }
}

// ---------------------------------------------------------------------------
// Main streaming kernel: 16 batch elements per wave-iteration.
//   D = WMMA(A2, Si_tile, WMMA(A1, Sr_tile, 0))
//   D rows 0-3 = Re(a), rows 4-7 = Im(a); columns = batch elements.
// C/D layout: lane n (0-15) holds D[0..7][n] in vgprs 0..7 -> the Z-weighted
// squared-magnitude reduction is 8 in-lane FMAs, then one coalesced store.
// B layout (f32 4x16, 2 VGPRs/lane): vgpr j, lane L holds
//   B[K][N] with K = j + (L<16 ? 0 : 2), N = L & 15  -> one b64 load/lane.
// ---------------------------------------------------------------------------
__global__ __launch_bounds__(256) void vqc_main(
    const float* __restrict__ sr, const float* __restrict__ si,
    const float* __restrict__ tab, float* __restrict__ out, int nTiles) {
  const int lane = threadIdx.x & 31;
  const int wave = threadIdx.x >> 5;
  const int wavesPerBlock = blockDim.x >> 5;
  const int gwave = blockIdx.x * wavesPerBlock + wave;
  const int nwaves = gridDim.x * wavesPerBlock;

  // Loop-invariant WMMA A operands.
  v2f a1, a2;
  a1.x = tab[lane];       a1.y = tab[32 + lane];
  a2.x = tab[64 + lane];  a2.y = tab[96 + lane];

  const int col = lane & 15;                 // batch element within tile
  const int koff = (lane < 16) ? 0 : 2;      // K half
  const bool writer = (lane < 16);

  for (int t = gwave; t < nTiles; t += nwaves) {
    const int base = t << 4;
    const int row = base + col;
    // Coalesced: 32 lanes cover one contiguous 256B tile of each input.
    v2f br = __builtin_nontemporal_load((const v2f*)(sr + row * 4 + koff));
    v2f bi = __builtin_nontemporal_load((const v2f*)(si + row * 4 + koff));

    v8f acc = {0.f, 0.f, 0.f, 0.f, 0.f, 0.f, 0.f, 0.f};
    acc = __builtin_amdgcn_wmma_f32_16x16x4_f32(false, a1, false, br,
                                                (short)0, acc, false, false);
    acc = __builtin_amdgcn_wmma_f32_16x16x4_f32(false, a2, false, bi,
                                                (short)0, acc, false, false);

    // out = z-weighted |a|^2, fully in-lane: 8 FMAs.
    float r0 = acc[0] * acc[0];
    r0 += acc[1] * acc[1];
    r0 -= acc[2] * acc[2];
    r0 -= acc[3] * acc[3];
    r0 += acc[4] * acc[4];
    r0 += acc[5] * acc[5];
    r0 -= acc[6] * acc[6];
    r0 -= acc[7] * acc[7];

    if (writer)
      __builtin_nontemporal_store(r0, out + base + lane);  // 64B coalesced
  }
}

// ---------------------------------------------------------------------------
extern "C" void kernel_launch(void* const* d_in, const int* in_sizes, int n_in,
                              void* d_out, int out_size, void* d_ws, size_t ws_size,
                              hipStream_t stream) {
  const float* ry = (const float*)d_in[0];
  const float* rz = (const float*)d_in[1];
  const float* sr = (const float*)d_in[2];
  const float* si = (const float*)d_in[3];
  float* out = (float*)d_out;
  float* tab = (float*)d_ws;  // 128 floats of scratch

  const int nTiles = out_size >> 4;  // BSZ / 16 (BSZ is a multiple of 16)

  vqc_build_tables<<<1, 32, 0, stream>>>(ry, rz, tab);
  vqc_main<<<1024, 256, 0, stream>>>(sr, si, tab, out, nTiles);
}